// EdgeConvGN_82721070120985
// MI455X (gfx1250) — compile-verified
//
#include <hip/hip_runtime.h>

typedef __attribute__((ext_vector_type(2))) float v2f;
typedef __attribute__((ext_vector_type(8))) float v8f;

// Problem dims from the reference setup.
constexpr int BB   = 2;
constexpr int C    = 128;      // C_in == C_out
constexpr int NPT  = 16384;    // N points
constexpr int KNN  = 16;       // K neighbours
constexpr float EPS_GN = 1e-5f;

// ---------------------------------------------------------------------------
// Kernel 0: zero the stats accumulator (B * 4 groups * {sum, sumsq} = 16 f32)
// ---------------------------------------------------------------------------
__global__ void zero_stats_kernel(float* __restrict__ stats) {
    if (threadIdx.x < 16) stats[threadIdx.x] = 0.0f;
}

// ---------------------------------------------------------------------------
// Kernel 1: WMMA f32 GEMM.
//   localT[b][n][c] = sum_k w1[c][k] * feature[b][k][n]
//   edgeT [b][n][c] = sum_k w2[c][k] * feature[b][k][n]
// Stored transposed (B,N,C) so later per-point channel gathers are contiguous.
// grid = (N/16, B), block = 256 (8 waves); wave w owns output rows [16w,16w+16).
// ---------------------------------------------------------------------------
__global__ __launch_bounds__(256) void gemm_wmma_kernel(
    const float* __restrict__ feat,   // (B, C, N)
    const float* __restrict__ w1,     // (C, C)
    const float* __restrict__ w2,     // (C, C)
    float* __restrict__ localT,       // (B, N, C)
    float* __restrict__ edgeT)        // (B, N, C)
{
    const int n0   = blockIdx.x * 16;
    const int b    = blockIdx.y;
    const int wave = threadIdx.x >> 5;
    const int lane = threadIdx.x & 31;
    const int c0   = wave * 16;           // output-channel tile base
    const int m    = lane & 15;           // M for A, N-column for B
    const int koff = (lane >> 4) * 2;     // K sub-pair: lanes 0-15 -> K{0,1}, 16-31 -> K{2,3}

    v8f acc1 = {};
    v8f acc2 = {};

    const float* fbase = feat + (size_t)b * C * NPT + (n0 + m);

    #pragma unroll
    for (int k = 0; k < C; k += 4) {
        // A tiles: row = c0+m of w1/w2, columns k+koff, k+koff+1 (float2, 8B aligned)
        v2f a1 = *(const v2f*)(w1 + (size_t)(c0 + m) * C + k + koff);
        v2f a2 = *(const v2f*)(w2 + (size_t)(c0 + m) * C + k + koff);
        // B tile: rows k+koff, k+koff+1 of feature, column n0+m (stride-N apart)
        v2f bt;
        bt.x = fbase[(size_t)(k + koff)     * NPT];
        bt.y = fbase[(size_t)(k + koff + 1) * NPT];
        acc1 = __builtin_amdgcn_wmma_f32_16x16x4_f32(false, a1, false, bt,
                                                     (short)0, acc1, false, false);
        acc2 = __builtin_amdgcn_wmma_f32_16x16x4_f32(false, a2, false, bt,
                                                     (short)0, acc2, false, false);
    }

    // C/D layout: VGPR r = M=r (lanes 0-15) / M=r+8 (lanes 16-31); N = lane&15.
    const int nn    = n0 + (lane & 15);
    const int cbase = c0 + (lane >> 4) * 8;
    float* lp = localT + ((size_t)b * NPT + nn) * C + cbase;
    float* ep = edgeT  + ((size_t)b * NPT + nn) * C + cbase;
    #pragma unroll
    for (int r = 0; r < 8; ++r) {
        lp[r] = acc1[r];
        ep[r] = acc2[r];
    }
}

// ---------------------------------------------------------------------------
// Kernel 2: group-norm statistics.
// Groups (64 channels each over C_total=256):
//   g0,g1: broadcast central  -> sum/sumsq of local, weighted by K
//   g2,g3: neighbour - central -> needs the KNN gather
// Thread t: channel c = t&127, n-phase = t>>7 (two n in flight per block).
// Per-thread accumulate, LDS tree-reduce, 8 atomicAdds per block.
// ---------------------------------------------------------------------------
__global__ __launch_bounds__(256) void gn_stats_kernel(
    const long long* __restrict__ knn,   // (B, N, K) int64
    const float* __restrict__ localT,    // (B, N, C)
    const float* __restrict__ edgeT,     // (B, N, C)
    float* __restrict__ stats,           // (B, 8)
    int nPerBlock)
{
    const int b       = blockIdx.y;
    const int n_start = blockIdx.x * nPerBlock;
    const int t       = threadIdx.x;
    const int c       = t & 127;
    const int half    = t >> 7;
    const int g       = c >> 6;   // 0/1 within the 128 channels

    float sc = 0.f, scq = 0.f, sd = 0.f, sdq = 0.f;

    for (int n = n_start + half; n < n_start + nPerBlock; n += 2) {
        const float lc = localT[((size_t)b * NPT + n) * C + c];
        sc  += lc;
        scq += lc * lc;
        const long long* ip = knn + ((size_t)b * NPT + n) * KNN;
        __builtin_prefetch(knn + ((size_t)b * NPT + n + 2) * KNN, 0, 0);
        #pragma unroll
        for (int k = 0; k < KNN; ++k) {
            const int idx = (int)ip[k];
            const float ev = edgeT[((size_t)b * NPT + idx) * C + c];
            const float d  = ev - lc;
            sd  += d;
            sdq += d * d;
        }
    }

    // stats layout per batch: [sum_g0, sq_g0, sum_g1, sq_g1, sum_g2, sq_g2, sum_g3, sq_g3]
    float vals[8] = {0.f, 0.f, 0.f, 0.f, 0.f, 0.f, 0.f, 0.f};
    vals[g * 2 + 0]     = sc  * (float)KNN;   // broadcast over K
    vals[g * 2 + 1]     = scq * (float)KNN;
    vals[4 + g * 2 + 0] = sd;
    vals[4 + g * 2 + 1] = sdq;

    __shared__ float red[256];
    for (int j = 0; j < 8; ++j) {
        red[t] = vals[j];
        __syncthreads();
        for (int s = 128; s > 0; s >>= 1) {
            if (t < s) red[t] += red[t + s];
            __syncthreads();
        }
        if (t == 0) atomicAdd(&stats[b * 8 + j], red[0]);
        __syncthreads();
    }
}

// ---------------------------------------------------------------------------
// Kernel 3: finalize — normalize, affine, ReLU, mean over K.
// Thread t = full channel (0..255). c<128: broadcast central (K-mean is identity).
// c>=128: re-gather neighbours, ReLU per k, average.
// ---------------------------------------------------------------------------
__global__ __launch_bounds__(256) void gn_out_kernel(
    const long long* __restrict__ knn,
    const float* __restrict__ localT,
    const float* __restrict__ edgeT,
    const float* __restrict__ stats,
    const float* __restrict__ gamma,     // (256)
    const float* __restrict__ beta,      // (256)
    float* __restrict__ out,             // (B, 256, N)
    int nPerBlock)
{
    const int b       = blockIdx.y;
    const int n_start = blockIdx.x * nPerBlock;
    const int cf      = threadIdx.x;     // 0..255 full channel
    const int g       = cf >> 6;         // group 0..3

    constexpr float invCount = 1.0f / (64.0f * (float)NPT * (float)KNN);

    __shared__ float smean[4], srs[4];
    if (threadIdx.x < 4) {
        const float s  = stats[b * 8 + threadIdx.x * 2];
        const float sq = stats[b * 8 + threadIdx.x * 2 + 1];
        const float mu = s * invCount;
        const float va = sq * invCount - mu * mu;
        smean[threadIdx.x] = mu;
        srs[threadIdx.x]   = rsqrtf(va + EPS_GN);
    }
    __syncthreads();

    const float mu = smean[g];
    const float rs = srs[g];
    const float ga = gamma[cf];
    const float be = beta[cf];

    if (cf < C) {
        const int c = cf;
        for (int n = n_start; n < n_start + nPerBlock; ++n) {
            const float x = localT[((size_t)b * NPT + n) * C + c];
            const float y = (x - mu) * rs * ga + be;
            out[((size_t)b * 2 * C + cf) * NPT + n] = fmaxf(y, 0.0f);
        }
    } else {
        const int c = cf - C;
        for (int n = n_start; n < n_start + nPerBlock; ++n) {
            const float lc = localT[((size_t)b * NPT + n) * C + c];
            const long long* ip = knn + ((size_t)b * NPT + n) * KNN;
            __builtin_prefetch(knn + ((size_t)b * NPT + n + 1) * KNN, 0, 0);
            float acc = 0.0f;
            #pragma unroll
            for (int k = 0; k < KNN; ++k) {
                const int idx = (int)ip[k];
                const float d = edgeT[((size_t)b * NPT + idx) * C + c] - lc;
                const float y = (d - mu) * rs * ga + be;
                acc += fmaxf(y, 0.0f);
            }
            out[((size_t)b * 2 * C + cf) * NPT + n] = acc * (1.0f / (float)KNN);
        }
    }
}

// ---------------------------------------------------------------------------
// Launcher
// ---------------------------------------------------------------------------
extern "C" void kernel_launch(void* const* d_in, const int* in_sizes, int n_in,
                              void* d_out, int out_size, void* d_ws, size_t ws_size,
                              hipStream_t stream) {
    const float*     feature = (const float*)d_in[0];      // (B, C, N)
    const long long* knn     = (const long long*)d_in[1];  // (B, N, K) int64
    const float*     w1      = (const float*)d_in[2];      // (C, C)
    const float*     w2      = (const float*)d_in[3];      // (C, C)
    const float*     gamma   = (const float*)d_in[4];      // (256)
    const float*     beta    = (const float*)d_in[5];      // (256)
    float*           out     = (float*)d_out;              // (B, 256, N)

    // Workspace: localT | edgeT | stats
    const size_t mat_elems = (size_t)BB * NPT * C;
    float* localT = (float*)d_ws;
    float* edgeT  = localT + mat_elems;
    float* stats  = edgeT + mat_elems;

    zero_stats_kernel<<<1, 32, 0, stream>>>(stats);

    dim3 gemm_grid(NPT / 16, BB);
    gemm_wmma_kernel<<<gemm_grid, 256, 0, stream>>>(feature, w1, w2, localT, edgeT);

    const int nPerBlockStats = 64;
    dim3 stats_grid(NPT / nPerBlockStats, BB);
    gn_stats_kernel<<<stats_grid, 256, 0, stream>>>(knn, localT, edgeT, stats, nPerBlockStats);

    const int nPerBlockOut = 64;
    dim3 out_grid(NPT / nPerBlockOut, BB);
    gn_out_kernel<<<out_grid, 256, 0, stream>>>(knn, localT, edgeT, stats,
                                                gamma, beta, out, nPerBlockOut);
}